// GCNEncoder_54288386621490
// MI455X (gfx1250) — compile-verified
//
#include <hip/hip_runtime.h>
#include <hip/hip_bf16.h>

#define D 64

typedef __attribute__((ext_vector_type(2))) float v2f;
typedef __attribute__((ext_vector_type(8))) float v8f;

// ---------------------------------------------------------------------------
// Relaxed agent-scope f32 atomic add -> should lower to global_atomic_add_f32
// ---------------------------------------------------------------------------
__device__ __forceinline__ void atomic_fadd(float* p, float v) {
    __hip_atomic_fetch_add(p, v, __ATOMIC_RELAXED, __HIP_MEMORY_SCOPE_AGENT);
}

// ---------------------------------------------------------------------------
// degree / dinv
// ---------------------------------------------------------------------------
__global__ void zero_f32(float* p, int n) {
    int i = blockIdx.x * blockDim.x + threadIdx.x;
    if (i < n) p[i] = 0.0f;
}

__global__ void deg_accum(const int* __restrict__ dst, float* __restrict__ deg, int E) {
    int e = blockIdx.x * blockDim.x + threadIdx.x;
    if (e < E) atomic_fadd(&deg[dst[e]], 1.0f);
}

__global__ void dinv_kernel(const float* __restrict__ deg, float* __restrict__ dinv, int n) {
    int i = blockIdx.x * blockDim.x + threadIdx.x;
    if (i < n) dinv[i] = __frsqrt_rn(deg[i] + 1.0f);
}

// ---------------------------------------------------------------------------
// Y = X @ W   (X: nrows x 64, W: 64 x 64, f32, exact)
// One wave computes a 16-row x 64-col strip with V_WMMA_F32_16X16X4_F32.
// A fragment (16x4):  lane<16: M=lane,   holds K=k0+0,k0+1
//                     lane>=16:M=lane-16,holds K=k0+2,k0+3
// B fragment (4x16):  lane<16: N=lane,   holds K=k0+0,k0+1
//                     lane>=16:N=lane-16,holds K=k0+2,k0+3
// C/D (16x16, 8 VGPR): VGPR r -> M=r (lanes 0-15) / M=r+8 (lanes 16-31), N=lane%16
// ---------------------------------------------------------------------------
__global__ void gemm_xw_wmma(const float* __restrict__ X,
                             const float* __restrict__ W,
                             float* __restrict__ Y, int nrows) {
    const int wave = (int)((blockIdx.x * blockDim.x + threadIdx.x) >> 5);
    const int lane = threadIdx.x & 31;
    const int row0 = wave * 16;
    if (row0 >= nrows) return;            // wave-uniform: EXEC stays all-1s for WMMA

    const int half = lane >> 4;           // 0 or 1
    const int l    = lane & 15;

    v8f acc0 = {}, acc1 = {}, acc2 = {}, acc3 = {};

    const float* xrow = X + (size_t)(row0 + l) * D;

#pragma unroll
    for (int k0 = 0; k0 < D; k0 += 4) {
        const int ka = k0 + 2 * half;
        v2f a;
        a.x = xrow[ka + 0];
        a.y = xrow[ka + 1];

        const float* w0 = W + (size_t)(ka + 0) * D;
        const float* w1 = W + (size_t)(ka + 1) * D;

        v2f b0; b0.x = w0[ 0 + l]; b0.y = w1[ 0 + l];
        v2f b1; b1.x = w0[16 + l]; b1.y = w1[16 + l];
        v2f b2; b2.x = w0[32 + l]; b2.y = w1[32 + l];
        v2f b3; b3.x = w0[48 + l]; b3.y = w1[48 + l];

        acc0 = __builtin_amdgcn_wmma_f32_16x16x4_f32(false, a, false, b0, (short)0, acc0, false, false);
        acc1 = __builtin_amdgcn_wmma_f32_16x16x4_f32(false, a, false, b1, (short)0, acc1, false, false);
        acc2 = __builtin_amdgcn_wmma_f32_16x16x4_f32(false, a, false, b2, (short)0, acc2, false, false);
        acc3 = __builtin_amdgcn_wmma_f32_16x16x4_f32(false, a, false, b3, (short)0, acc3, false, false);
    }

#pragma unroll
    for (int r = 0; r < 8; ++r) {
        const int row = row0 + r + 8 * half;
        float* yrow = Y + (size_t)row * D + l;
        yrow[ 0] = acc0[r];
        yrow[16] = acc1[r];
        yrow[32] = acc2[r];
        yrow[48] = acc3[r];
    }
}

// ---------------------------------------------------------------------------
// agg[i] = y[i] * dinv[i]^2 + b   (self-loop + bias; initializes scatter target)
// one thread per (node, 4 floats)
// ---------------------------------------------------------------------------
__global__ void self_bias(const float* __restrict__ Y,
                          const float* __restrict__ dinv,
                          const float* __restrict__ b,
                          float* __restrict__ agg, int n) {
    int t = blockIdx.x * blockDim.x + threadIdx.x;
    int i = t >> 4;
    int q = (t & 15) * 4;
    if (i >= n) return;
    float di = dinv[i];
    float c  = di * di;
    const float4 y  = *(const float4*)(Y + (size_t)i * D + q);
    const float4 bb = *(const float4*)(b + q);
    float4 r;
    r.x = fmaf(y.x, c, bb.x);
    r.y = fmaf(y.y, c, bb.y);
    r.z = fmaf(y.z, c, bb.z);
    r.w = fmaf(y.w, c, bb.w);
    *(float4*)(agg + (size_t)i * D + q) = r;
}

// ---------------------------------------------------------------------------
// agg[dst] += y[src] * dinv[src]*dinv[dst]   (16 threads per edge, float4 each)
// ---------------------------------------------------------------------------
__global__ void edge_scatter(const float* __restrict__ Y,
                             const int* __restrict__ src,
                             const int* __restrict__ dst,
                             const float* __restrict__ dinv,
                             float* __restrict__ agg, int E) {
    int t = blockIdx.x * blockDim.x + threadIdx.x;
    int e = t >> 4;
    int q = (t & 15) * 4;
    if (e >= E) return;
    int s = src[e];
    int d = dst[e];
    float coef = dinv[s] * dinv[d];
    const float4 v = *(const float4*)(Y + (size_t)s * D + q);
    float* out = agg + (size_t)d * D + q;
    atomic_fadd(out + 0, v.x * coef);
    atomic_fadd(out + 1, v.y * coef);
    atomic_fadd(out + 2, v.z * coef);
    atomic_fadd(out + 3, v.w * coef);
}

__global__ void relu_inplace(float* __restrict__ x, int n4) {
    int i = blockIdx.x * blockDim.x + threadIdx.x;
    if (i >= n4) return;
    float4 v = ((float4*)x)[i];
    v.x = fmaxf(v.x, 0.0f);
    v.y = fmaxf(v.y, 0.0f);
    v.z = fmaxf(v.z, 0.0f);
    v.w = fmaxf(v.w, 0.0f);
    ((float4*)x)[i] = v;
}

// ---------------------------------------------------------------------------
// host-side launch
// ---------------------------------------------------------------------------
static inline int cdiv(long long a, int b) { return (int)((a + b - 1) / b); }

static void gcn_layer(const float* x, const float* W, const float* b,
                      const int* src, const int* dst, const float* dinv,
                      float* y, float* agg, int N, int E, hipStream_t stream) {
    const int waves  = (N + 15) / 16;
    const int gblk   = cdiv(waves, 8);          // 8 waves (256 threads) per block
    gemm_xw_wmma<<<gblk, 256, 0, stream>>>(x, W, y, N);
    self_bias<<<cdiv((long long)N * 16, 256), 256, 0, stream>>>(y, dinv, b, agg, N);
    edge_scatter<<<cdiv((long long)E * 16, 256), 256, 0, stream>>>(y, src, dst, dinv, agg, E);
}

extern "C" void kernel_launch(void* const* d_in, const int* in_sizes, int n_in,
                              void* d_out, int out_size, void* d_ws, size_t ws_size,
                              hipStream_t stream) {
    const int N = in_sizes[0] / D;      // 50000
    const int E = in_sizes[1] / 2;      // 800000

    const float* x0  = (const float*)d_in[0];
    const int*   ei  = (const int*)d_in[1];
    const int*   src = ei;
    const int*   dst = ei + E;
    const float* W1  = (const float*)d_in[2];
    const float* b1  = (const float*)d_in[3];
    const float* W2  = (const float*)d_in[4];
    const float* b2  = (const float*)d_in[5];
    const float* Ws  = (const float*)d_in[6];
    const float* bs  = (const float*)d_in[7];
    const float* Wmu = (const float*)d_in[8];
    const float* bmu = (const float*)d_in[9];
    const float* Wlv = (const float*)d_in[10];
    const float* blv = (const float*)d_in[11];

    // workspace partition
    float* deg  = (float*)d_ws;
    float* dinv = deg + N;
    float* bufA = dinv + N;
    float* bufB = bufA + (size_t)N * D;
    float* bufY = bufB + (size_t)N * D;

    float* mu = (float*)d_out;
    float* lv = mu + (size_t)N * D;

    // degrees (with self-loop) and dinv
    zero_f32<<<cdiv(N, 256), 256, 0, stream>>>(deg, N);
    deg_accum<<<cdiv(E, 256), 256, 0, stream>>>(dst, deg, E);
    dinv_kernel<<<cdiv(N, 256), 256, 0, stream>>>(deg, dinv, N);

    // encoder layers
    gcn_layer(x0,   W1, b1, src, dst, dinv, bufY, bufA, N, E, stream);   // x1 in bufA
    gcn_layer(bufA, W2, b2, src, dst, dinv, bufY, bufB, N, E, stream);   // x2 in bufB
    gcn_layer(bufB, Ws, bs, src, dst, dinv, bufY, bufA, N, E, stream);   // x3 in bufA
    relu_inplace<<<cdiv((long long)N * D / 4, 256), 256, 0, stream>>>(bufA, N * D / 4);

    // heads -> d_out directly
    gcn_layer(bufA, Wmu, bmu, src, dst, dinv, bufY, mu, N, E, stream);
    gcn_layer(bufA, Wlv, blv, src, dst, dinv, bufY, lv, N, E, stream);
}